// DGG_LearnableK_SDD_47536698032964
// MI455X (gfx1250) — compile-verified
//
#include <hip/hip_runtime.h>
#include <hip/hip_bf16.h>

typedef float v2f __attribute__((ext_vector_type(2)));
typedef float v8f __attribute__((ext_vector_type(8)));

#define NN        4096          // nodes per batch
#define IN_DIM    32
#define LATENT    64
#define ROWS      8192          // B * NN
#define ADJ_ELEMS 33554432ull   // 2*4096*4096
#define TPB       512

// ---------------------------------------------------------------------------
// Kernel 1: k-path.  k = relu(x@W1+b1)@W2+b2 @ Wkp + bkp + 1
// One wave per 16 rows; f32 WMMA 16x16x4; 4 waves per block (64 rows/block).
// ---------------------------------------------------------------------------
__global__ __launch_bounds__(128) void knet_wmma_kernel(
    const float* __restrict__ x,     // [8192,32]
    const float* __restrict__ W1,    // [32,64]
    const float* __restrict__ b1,    // [64]
    const float* __restrict__ W2,    // [64,64]
    const float* __restrict__ b2,    // [64]
    const float* __restrict__ Wkp,   // [64]
    const float* __restrict__ bkp,   // [1]
    float* __restrict__ kOut)        // [8192]
{
    __shared__ float h1s[4][16 * LATENT];   // per-wave relu(x@W1+b1) staging
    __shared__ float reds[4][16][16];       // per-wave k reduction scratch

    const int lane = threadIdx.x & 31;
    const int wave = threadIdx.x >> 5;
    const int rowBase = (blockIdx.x * 4 + wave) * 16;

    const int m  = lane & 15;   // M for A-frag / N for B-frag
    const int kh = lane >> 4;   // K half selector (K pair 0/1 vs 2/3)

    // ---------------- GEMM1: [16,32] x [32,64] + bias, relu -> LDS ----------
    v8f acc[4];
#pragma unroll
    for (int nt = 0; nt < 4; ++nt) acc[nt] = (v8f){0.f,0.f,0.f,0.f,0.f,0.f,0.f,0.f};

#pragma unroll
    for (int kk = 0; kk < 8; ++kk) {
        const int k0 = kk * 4 + 2 * kh;
        v2f a;
        const float* xr = x + (size_t)(rowBase + m) * IN_DIM + k0;
        a.x = xr[0];
        a.y = xr[1];
#pragma unroll
        for (int nt = 0; nt < 4; ++nt) {
            const int n = nt * 16 + m;
            v2f bf;
            bf.x = W1[(size_t)k0 * LATENT + n];
            bf.y = W1[(size_t)(k0 + 1) * LATENT + n];
            acc[nt] = __builtin_amdgcn_wmma_f32_16x16x4_f32(
                false, a, false, bf, (short)0, acc[nt], false, false);
        }
    }

#pragma unroll
    for (int nt = 0; nt < 4; ++nt) {
        const int n = nt * 16 + m;
        const float bias = b1[n];
#pragma unroll
        for (int r = 0; r < 8; ++r) {
            const int M = r + 8 * kh;
            float v = acc[nt][r] + bias;
            h1s[wave][M * LATENT + n] = v > 0.f ? v : 0.f;
        }
    }
    __syncthreads();

    // ---------------- GEMM2: [16,64] x [64,64] ------------------------------
    v8f acc2[4];
#pragma unroll
    for (int nt = 0; nt < 4; ++nt) acc2[nt] = (v8f){0.f,0.f,0.f,0.f,0.f,0.f,0.f,0.f};

#pragma unroll
    for (int kk = 0; kk < 16; ++kk) {
        const int k0 = kk * 4 + 2 * kh;
        v2f a;
        a.x = h1s[wave][m * LATENT + k0];
        a.y = h1s[wave][m * LATENT + k0 + 1];
#pragma unroll
        for (int nt = 0; nt < 4; ++nt) {
            const int n = nt * 16 + m;
            v2f bf;
            bf.x = W2[(size_t)k0 * LATENT + n];
            bf.y = W2[(size_t)(k0 + 1) * LATENT + n];
            acc2[nt] = __builtin_amdgcn_wmma_f32_16x16x4_f32(
                false, a, false, bf, (short)0, acc2[nt], false, false);
        }
    }

    // ---------------- k = (h2 + b2) . Wkp + bkp + 1 -------------------------
    float psum[8];
#pragma unroll
    for (int r = 0; r < 8; ++r) psum[r] = 0.f;
#pragma unroll
    for (int nt = 0; nt < 4; ++nt) {
        const int n = nt * 16 + m;
        const float wk = Wkp[n];
        const float bb = b2[n];
#pragma unroll
        for (int r = 0; r < 8; ++r)
            psum[r] += (acc2[nt][r] + bb) * wk;
    }
#pragma unroll
    for (int r = 0; r < 8; ++r) {
        const int M = r + 8 * kh;
        reds[wave][M][m] = psum[r];
    }
    __syncthreads();

    if (lane < 16) {
        float s = 0.f;
#pragma unroll
        for (int c = 0; c < 16; ++c) s += reds[wave][lane][c];
        kOut[rowBase + lane] = s + bkp[0] + 1.0f;
    }
}

// ---------------------------------------------------------------------------
// Kernel 2: per-row gumbel + stable descending rank (bitonic) + sigmoid gate.
// One block (512 threads) per row of 4096 columns.  48 KB LDS.
// Key = (~asc_float_key(ep) << 32) | column  -> ascending sort == descending
// by value, ties broken by ascending column (matches stable argsort(-ep)).
// ---------------------------------------------------------------------------
__global__ __launch_bounds__(TPB) void gumbel_rank_kernel(
    const float* __restrict__ gumbel,   // [8192,4096]
    const float* __restrict__ kvals,    // [8192]
    float* __restrict__ adj)            // [8192,4096]
{
    __shared__ unsigned long long keys[NN];
    __shared__ float outv[NN];

    const int row  = blockIdx.x;            // 0..8191
    const int diag = row & (NN - 1);        // i = row % 4096
    const int tid  = threadIdx.x;
    const float* __restrict__ grow = gumbel + (size_t)row * NN;

    // 1) edge_prob + key packing
#pragma unroll
    for (int e = 0; e < NN / TPB; ++e) {
        const int j = tid + e * TPB;
        const float u = __builtin_nontemporal_load(&grow[j]);
        float ep;
        if (j == diag) {
            ep = 0.0f;                       // mask zeroes the diagonal noise
        } else {
            ep = -logf(-logf(u + 1e-20f) + 1e-20f);
        }
        const unsigned int bits = __float_as_uint(ep);
        const unsigned int akey = (bits & 0x80000000u) ? ~bits : (bits | 0x80000000u);
        const unsigned int dkey = ~akey;     // ascending dkey == descending ep
        keys[j] = ((unsigned long long)dkey << 32) | (unsigned int)j;
    }
    __syncthreads();

    // 2) bitonic sort, ascending on 64-bit key
    for (int k = 2; k <= NN; k <<= 1) {
        for (int j = k >> 1; j > 0; j >>= 1) {
#pragma unroll
            for (int t = tid; t < NN / 2; t += TPB) {
                const int i  = ((t & ~(j - 1)) << 1) | (t & (j - 1));
                const int ip = i | j;
                const unsigned long long a = keys[i];
                const unsigned long long b = keys[ip];
                const bool up = ((i & k) == 0);
                if ((a > b) == up) { keys[i] = b; keys[ip] = a; }
            }
            __syncthreads();
        }
    }

    // 3) decode value, apply first_k gate at sorted position, scatter in LDS
    const float kv    = kvals[row];
    const float shift = (kv - 1.0f) * 7.0f;
#pragma unroll
    for (int e = 0; e < NN / TPB; ++e) {
        const int r = tid + e * TPB;
        const unsigned long long key = keys[r];
        const int j = (int)(unsigned int)key;
        const unsigned int akey = ~(unsigned int)(key >> 32);
        const unsigned int bits = (akey & 0x80000000u) ? (akey & 0x7FFFFFFFu) : ~akey;
        const float val = __uint_as_float(bits);
        const float z   = 2.0f - 7.0f * (float)r + shift;   // x_support[r] + shift
        const float sig = 1.0f / (1.0f + expf(-z));
        outv[j] = val * sig;
    }
    __syncthreads();

    // 4) coalesced streaming store of the row
    float* __restrict__ arow = adj + (size_t)row * NN;
#pragma unroll
    for (int e = 0; e < NN / TPB; ++e) {
        const int c = tid + e * TPB;
        __builtin_nontemporal_store(outv[c], &arow[c]);
    }
}

// ---------------------------------------------------------------------------
extern "C" void kernel_launch(void* const* d_in, const int* in_sizes, int n_in,
                              void* d_out, int out_size, void* d_ws, size_t ws_size,
                              hipStream_t stream) {
    (void)in_sizes; (void)n_in; (void)out_size; (void)d_ws; (void)ws_size;

    const float* x      = (const float*)d_in[0];
    // d_in[1] = temp (unused by reference)
    const float* gumbel = (const float*)d_in[2];
    // d_in[3..6] = W_proj, b_proj, w_dist, b_dist: dead code (log_p == 0 exactly)
    const float* W1  = (const float*)d_in[7];
    const float* b1  = (const float*)d_in[8];
    const float* W2  = (const float*)d_in[9];
    const float* b2  = (const float*)d_in[10];
    const float* Wkp = (const float*)d_in[11];
    const float* bkp = (const float*)d_in[12];

    float* adj  = (float*)d_out;              // [2,4096,4096]
    float* kOut = (float*)d_out + ADJ_ELEMS;  // [2,4096,1]

    // k-path first (kernel 2 consumes kOut; same stream => ordered)
    knet_wmma_kernel<<<ROWS / 64, 128, 0, stream>>>(x, W1, b1, W2, b2, Wkp, bkp, kOut);
    gumbel_rank_kernel<<<ROWS, TPB, 0, stream>>>(gumbel, kOut, adj);
}